// QuantumDecoder_69939247448632
// MI455X (gfx1250) — compile-verified
//
#include <hip/hip_runtime.h>
#include <math.h>

typedef float v2f __attribute__((ext_vector_type(2)));
typedef float v8f __attribute__((ext_vector_type(8)));

// D = A(16x4 f32) * B(4x16 f32) + C(16x16 f32)
#define WMMA_F32X4(A, Bv, C) \
    __builtin_amdgcn_wmma_f32_16x16x4_f32(false, (A), false, (Bv), (short)0, (C), false, false)

#define LATD 128
#define OUTD 784
#define HD   8
#define NT   49          // 784 / 16 N-tiles

__device__ __forceinline__ float fast_tanh(float x) {
#if __has_builtin(__builtin_amdgcn_tanhf)
    return __builtin_amdgcn_tanhf(x);        // v_tanh_f32 (gfx1250)
#else
    return tanhf(x);
#endif
}

__device__ __forceinline__ float fast_sigmoid(float x) {
#if __has_builtin(__builtin_amdgcn_tanhf)
    return __builtin_amdgcn_tanhf(x * 0.5f) * 0.5f + 0.5f;   // 3 ops, 1 TRANS
#else
    return __builtin_amdgcn_rcpf(1.0f + __builtin_amdgcn_exp2f(x * -1.44269504f));
#endif
}

__launch_bounds__(256)
__global__ void qdec_fused(const float* __restrict__ z,
                           const float* __restrict__ W_in,
                           const float* __restrict__ b_in,
                           const float* __restrict__ theta,
                           const float* __restrict__ W1,
                           const float* __restrict__ b1,
                           const float* __restrict__ W2,
                           const float* __restrict__ b2,
                           float* __restrict__ out)
{
    // Pre-swizzled per-lane WMMA B-fragments (no divergence in hot loops)
    __shared__ float sBf1[32 * 32 * 2];   // [kchunk][lane][2]  W_in frags, 8 KB
    __shared__ float sBf2[NT * 32 * 4];   // [ntile][lane][4]   W2 frags,  25 KB
    __shared__ float sb2[OUTD];           // 3136 B
    __shared__ float sLat[8][32][4];      // 4096 B
    __shared__ float sH[8][32][HD];       // 8192 B

    const int tid = threadIdx.x;

    // ---- stage W_in fragments: B is 4x16, rows kr..kr+1 per vgpr half -------
    for (int i = tid; i < 32 * 32; i += 256) {
        const int c  = i >> 5;
        const int ln = i & 31;
        const int hf = ln >> 4;
        const int cl = ln & 15;
        const int kr = c * 4 + 2 * hf;
        sBf1[i * 2 + 0] = (cl < 4) ? W_in[kr * 4 + cl]       : 0.0f;
        sBf1[i * 2 + 1] = (cl < 4) ? W_in[(kr + 1) * 4 + cl] : 0.0f;
    }
    // ---- stage W2 fragments: rows {2h,2h+1} (k-chunk 0) and {4+2h,5+2h} -----
    for (int i = tid; i < NT * 32; i += 256) {
        const int t  = i >> 5;
        const int ln = i & 31;
        const int hf = ln >> 4;
        const int cl = ln & 15;
        const int col = t * 16 + cl;
        sBf2[i * 4 + 0] = W2[(2 * hf) * OUTD + col];
        sBf2[i * 4 + 1] = W2[(2 * hf + 1) * OUTD + col];
        sBf2[i * 4 + 2] = W2[(4 + 2 * hf) * OUTD + col];
        sBf2[i * 4 + 3] = W2[(5 + 2 * hf) * OUTD + col];
    }
    for (int i = tid; i < OUTD; i += 256) sb2[i] = b2[i];
    __syncthreads();

    const int wave = tid >> 5;
    const int lane = tid & 31;
    const int half = lane >> 4;     // 0: lanes 0-15, 1: lanes 16-31
    const int l16  = lane & 15;
    const int rowBase = blockIdx.x * 256 + wave * 32;

    // ---------------- stage 1: lat_pre = z @ W_in  (f32 WMMA, K=128) --------
    v8f c0 = {}; v8f c1 = {};
    const float* zrow0 = z + (size_t)(rowBase + l16)      * LATD + 2 * half;
    const float* zrow1 = z + (size_t)(rowBase + 16 + l16) * LATD + 2 * half;
    #pragma unroll 8
    for (int c = 0; c < 32; ++c) {
        float2 t0 = *(const float2*)(zrow0 + 4 * c);   // A: K = 4c+2*half, +1
        float2 t1 = *(const float2*)(zrow1 + 4 * c);
        v2f bf = *(const v2f*)&sBf1[(c * 32 + lane) * 2];
        v2f a0; a0.x = t0.x; a0.y = t0.y;
        v2f a1; a1.x = t1.x; a1.y = t1.y;
        c0 = WMMA_F32X4(a0, bf, c0);
        c1 = WMMA_F32X4(a1, bf, c1);
    }

    // transpose C fragment (cols 0..3 valid) so lane L owns batch row L
    if (l16 < 4) {
        #pragma unroll
        for (int v = 0; v < 8; ++v) {
            sLat[wave][v + 8 * half][l16]      = c0[v];   // M = v (+8 upper half)
            sLat[wave][16 + v + 8 * half][l16] = c1[v];
        }
    }
    __syncthreads();

    float ang[4];
    #pragma unroll
    for (int q = 0; q < 4; ++q)
        ang[q] = fast_tanh(sLat[wave][lane][q] + b_in[q]);

    // ---------------- stage 2: 4-qubit statevector sim (registers) ----------
    float sr[16], si[16];
    {
        float cq[4], sq[4];
        #pragma unroll
        for (int q = 0; q < 4; ++q) {
            float s_, c_; __sincosf(0.5f * ang[q], &s_, &c_);
            sq[q] = s_; cq[q] = c_;
        }
        // RY(ang) on |0000> -> real product state, Gray-factored
        float p01[4], p23[4];
        p01[0] = cq[0] * cq[1]; p01[1] = sq[0] * cq[1];
        p01[2] = cq[0] * sq[1]; p01[3] = sq[0] * sq[1];
        p23[0] = cq[2] * cq[3]; p23[1] = sq[2] * cq[3];
        p23[2] = cq[2] * sq[3]; p23[3] = sq[2] * sq[3];
        #pragma unroll
        for (int i = 0; i < 16; ++i) { sr[i] = p01[i & 3] * p23[i >> 2]; si[i] = 0.0f; }
    }

    #pragma unroll
    for (int layer = 0; layer < 2; ++layer) {
        #pragma unroll
        for (int q = 0; q < 4; ++q) {
            const float th = theta[layer * 4 + q];      // uniform -> scalar load
            float s_, c_; __sincosf(0.5f * th, &s_, &c_);
            // Fused U = RZ*RY*RX = [[P+iQ, -R+iQ],[R+iQ, P-iQ]]
            const float cs = c_ * s_;
            const float P  = (c_ + s_) * (1.0f - cs);   // c^3 + s^3
            const float Q  = cs * (s_ - c_);
            const float R  = cs * (c_ + s_);
            const int step = 1 << q;
            #pragma unroll
            for (int i = 0; i < 16; ++i) {
                if (i & step) continue;
                const int j = i | step;
                const float ar = sr[i], ai = si[i], br = sr[j], bi = si[j];
                sr[i] = P * ar - Q * ai - R * br - Q * bi;
                si[i] = Q * ar + P * ai + Q * br - R * bi;
                sr[j] = R * ar - Q * ai + P * br + Q * bi;
                si[j] = Q * ar + R * ai - Q * br + P * bi;
            }
        }
        // CX chain (0,1),(1,2),(2,3),(3,0) as index permutations
        #pragma unroll
        for (int e = 0; e < 4; ++e) {
            const int ctl = e, tgt = (e + 1) & 3;
            float tr[16], ti[16];
            #pragma unroll
            for (int m = 0; m < 16; ++m) {
                const int src = m ^ (((m >> ctl) & 1) << tgt);
                tr[m] = sr[src]; ti[m] = si[src];
            }
            #pragma unroll
            for (int m = 0; m < 16; ++m) { sr[m] = tr[m]; si[m] = ti[m]; }
        }
    }

    // Z expectations, reversed column order
    float zq[4] = {0.f, 0.f, 0.f, 0.f};
    #pragma unroll
    for (int i = 0; i < 16; ++i) {
        const float p = sr[i] * sr[i] + si[i] * si[i];
        #pragma unroll
        for (int q = 0; q < 4; ++q) zq[q] += ((i >> q) & 1) ? -p : p;
    }
    float qe[4];
    #pragma unroll
    for (int q = 0; q < 4; ++q) qe[q] = zq[3 - q];

    // ---------------- stage 3: h = relu(qexp @ W1 + b1) ----------------------
    #pragma unroll
    for (int k = 0; k < HD; ++k) {
        float acc = b1[k];
        #pragma unroll
        for (int q = 0; q < 4; ++q) acc += qe[q] * W1[q * HD + k];
        sH[wave][lane][k] = fmaxf(acc, 0.0f);
    }
    __syncthreads();

    // ---------------- stage 4: out = sigmoid(h @ W2 + b2)  (f32 WMMA, K=8) ---
    // A fragments invariant across all 49 N-tiles.
    v2f a00, a04, a10, a14;
    a00.x = sH[wave][l16][2 * half];           a00.y = sH[wave][l16][2 * half + 1];
    a04.x = sH[wave][l16][4 + 2 * half];       a04.y = sH[wave][l16][4 + 2 * half + 1];
    a10.x = sH[wave][16 + l16][2 * half];      a10.y = sH[wave][16 + l16][2 * half + 1];
    a14.x = sH[wave][16 + l16][4 + 2 * half];  a14.y = sH[wave][16 + l16][4 + 2 * half + 1];

    for (int t = 0; t < NT; ++t) {
        const float4 bq = *(const float4*)&sBf2[(t * 32 + lane) * 4];  // ds_load_b128
        v2f b0; b0.x = bq.x; b0.y = bq.y;
        v2f b4; b4.x = bq.z; b4.y = bq.w;

        const int col = t * 16 + l16;
        const float bias = sb2[col];
        v8f d0 = {bias, bias, bias, bias, bias, bias, bias, bias};  // fold b2 into C
        v8f d1 = d0;
        d0 = WMMA_F32X4(a00, b0, d0);
        d0 = WMMA_F32X4(a04, b4, d0);
        d1 = WMMA_F32X4(a10, b0, d1);
        d1 = WMMA_F32X4(a14, b4, d1);

        float* o0 = out + (size_t)(rowBase + 8 * half) * OUTD + col;
        float* o1 = out + (size_t)(rowBase + 16 + 8 * half) * OUTD + col;
        #pragma unroll
        for (int v = 0; v < 8; ++v) {
            o0[(size_t)v * OUTD] = fast_sigmoid(d0[v]);
            o1[(size_t)v * OUTD] = fast_sigmoid(d1[v]);
        }
    }
}

extern "C" void kernel_launch(void* const* d_in, const int* in_sizes, int n_in,
                              void* d_out, int out_size, void* d_ws, size_t ws_size,
                              hipStream_t stream) {
    const float* z     = (const float*)d_in[0];
    const float* W_in  = (const float*)d_in[1];
    const float* b_in  = (const float*)d_in[2];
    const float* theta = (const float*)d_in[3];
    const float* W1    = (const float*)d_in[4];
    const float* b1    = (const float*)d_in[5];
    const float* W2    = (const float*)d_in[6];
    const float* b2    = (const float*)d_in[7];
    float* out = (float*)d_out;

    const int B = in_sizes[0] / LATD;   // 65536
    dim3 grid(B / 256), block(256);     // 8 waves/block, 32 batch rows/wave
    qdec_fused<<<grid, block, 0, stream>>>(z, W_in, b_in, theta, W1, b1, W2, b2, out);
}